// NElectronSlater_80229989090023
// MI455X (gfx1250) — compile-verified
//
#include <hip/hip_runtime.h>

// One wave32 per walker. Matrix built via V_WMMA_F32_16X16X4_F32 (rank-1
// outer product, K=1 live) directly into the 16x16 f32 C/D fragment layout:
//   lane l (col = l&15, hi = l>>4) holds A[r + 8*hi][col] in mat[r], r=0..7.
// Then unpivoted LU entirely in registers using ds_bpermute broadcasts,
// accumulating sum(log|pivot|); output = 2 * logabsdet.
// v1 -> v2: replace IEEE 1.0f/p (div_scale/div_fmas/div_fixup chain, ~10 VALU
// per step) with single v_rcp_f32 via __builtin_amdgcn_rcpf.

typedef __attribute__((ext_vector_type(2))) float v2f;
typedef __attribute__((ext_vector_type(8))) float v8f;

__device__ __forceinline__ float lane_bcast(float v, int srcLane) {
    return __int_as_float(__builtin_amdgcn_ds_bpermute(srcLane << 2, __float_as_int(v)));
}

__global__ __launch_bounds__(256) void
NElectronSlater_80229989090023_kernel(const float* __restrict__ x,
                                      const float* __restrict__ alpha,
                                      float* __restrict__ out,
                                      int nWalkers) {
    const int tid  = blockIdx.x * blockDim.x + threadIdx.x;
    const int wave = tid >> 5;
    const int lane = threadIdx.x & 31;
    if (wave >= nWalkers) return;          // wave-uniform: EXEC stays all-ones

    const int col  = lane & 15;            // column owned by this lane
    const int hi16 = lane & 16;            // 0 (rows 0-7) or 16 (rows 8-15)

    // ---- Build fragments for D = (-x) x alpha via WMMA f32 16x16x4 ----
    // A (16x4): lanes 0-15 VGPR0 hold K=0 -> -x[M]; everything else zero.
    // B (4x16): lanes 0-15 VGPR0 hold K=0 row -> alpha[N]; everything else zero.
    float xl = x[(size_t)wave * 16 + col]; // always in-bounds (clamped by &15)
    float al = alpha[col];
    v2f afrag, bfrag;
    afrag.x = (hi16 == 0) ? -xl : 0.0f;
    afrag.y = 0.0f;
    bfrag.x = (hi16 == 0) ?  al : 0.0f;
    bfrag.y = 0.0f;

    v8f c = {};
    // 8 args: (neg_a, A, neg_b, B, c_mod, C, reuse_a, reuse_b)
    c = __builtin_amdgcn_wmma_f32_16x16x4_f32(false, afrag, false, bfrag,
                                              (short)0, c, false, false);

    // Slater matrix: elementwise exp on the fragment
    float mat[8];
#pragma unroll
    for (int r = 0; r < 8; ++r) mat[r] = __expf(c[r]);

    // ---- In-register LU (no pivoting), accumulate log|diag| ----
    float acc = 0.0f;
#pragma unroll
    for (int k = 0; k < 16; ++k) {
        const int kr = k & 7;              // VGPR slot of pivot row
        const int kb = (k >> 3) << 4;      // lane-group base holding row k
        float p    = lane_bcast(mat[kr], kb + k);    // pivot A[k][k] (uniform)
        float rk   = lane_bcast(mat[kr], kb + col);  // A[k][col] for my column
        float invp = __builtin_amdgcn_rcpf(p);       // single v_rcp_f32
        acc += __logf(fabsf(p));
#pragma unroll
        for (int r = 0; r < 8; ++r) {
            if (r + 8 > k) {               // compile-time prune: some hi may update
                // A[gr][k] with gr = r + 8*hi lives in lane (hi*16 + k), slot r
                float ck  = lane_bcast(mat[r], hi16 + k);
                float upd = fmaf(-(ck * invp), rk, mat[r]);
                mat[r] = ((r + (hi16 >> 1)) > k) ? upd : mat[r];  // gr > k
            }
        }
    }

    if (lane == 0) out[wave] = 2.0f * acc; // log|psi|^2
}

extern "C" void kernel_launch(void* const* d_in, const int* in_sizes, int n_in,
                              void* d_out, int out_size, void* d_ws, size_t ws_size,
                              hipStream_t stream) {
    const float* x     = (const float*)d_in[0];  // [B,16] f32
    const float* alpha = (const float*)d_in[1];  // [16]   f32
    float*       out   = (float*)d_out;          // [B]    f32

    const int B = in_sizes[0] / 16;              // 131072
    const int wavesPerBlock = 8;                 // 256 threads = 8 wave32
    const int blocks = (B + wavesPerBlock - 1) / wavesPerBlock;

    NElectronSlater_80229989090023_kernel<<<blocks, 256, 0, stream>>>(x, alpha, out, B);
}